// GraphTransformer_43568148250688
// MI455X (gfx1250) — compile-verified
//
#include <hip/hip_runtime.h>
#include <math.h>

#define DHID 25
#define HPAD 28          // K-padded row stride for h buffers (multiple of 4)
#define HEADS 5
#define CH 5
#define KPAIRS 14        // 28/2 k-pairs in swizzled LDS weights
#define NPADW 32         // padded N dimension in swizzled LDS weights
#define WSTRIDE (KPAIRS * 2 * NPADW)   // 896 floats per weight matrix in LDS

typedef __attribute__((ext_vector_type(2))) float v2f;
typedef __attribute__((ext_vector_type(8))) float v8f;

// ---------- helpers: order-preserving float<->uint key for atomic max ----------
__device__ __forceinline__ unsigned fkey(float f) {
    unsigned u = __float_as_uint(f);
    return (u & 0x80000000u) ? ~u : (u | 0x80000000u);
}
__device__ __forceinline__ float fdecode(unsigned kk) {
    unsigned u = (kk & 0x80000000u) ? (kk & 0x7FFFFFFFu) : ~kk;
    return __uint_as_float(u);
}

// ---------- input projection: h = x @ Wi + bi  (N x 6 -> Npad x HPAD, padded) ----------
__global__ void __launch_bounds__(256)
input_proj_kernel(const float* __restrict__ x, const float* __restrict__ Wi,
                  const float* __restrict__ bi, float* __restrict__ h, int N) {
    __shared__ float sWi[6 * DHID];
    __shared__ float sBi[DHID];
    if (threadIdx.x < 6 * DHID) sWi[threadIdx.x] = Wi[threadIdx.x];
    if (threadIdx.x < DHID) sBi[threadIdx.x] = bi[threadIdx.x];
    __syncthreads();
    for (int n = blockIdx.x * 256 + threadIdx.x; n < N; n += gridDim.x * 256) {
        float xv[6];
        #pragma unroll
        for (int i = 0; i < 6; ++i) xv[i] = x[n * 6 + i];
        #pragma unroll
        for (int j = 0; j < DHID; ++j) {
            float acc = sBi[j];
            #pragma unroll
            for (int i = 0; i < 6; ++i) acc += xv[i] * sWi[i * DHID + j];
            h[n * HPAD + j] = acc;
        }
    }
}

// ---------- fused q/k/v/skip projection via V_WMMA_F32_16X16X4_F32 ----------
// All inner-loop loads are unconditional:
//   A: one aligned v2f global load from K-padded h (stride HPAD)
//   B: one ds_load_b64 from zero-padded, pair-swizzled LDS weights
// One wave per 16-node tile; 8 v8f accumulators (q/k/v/skip x 2 n-tiles).
__global__ void __launch_bounds__(256)
qkvs_wmma_kernel(const float* __restrict__ h,
                 const float* __restrict__ Wq, const float* __restrict__ bq,
                 const float* __restrict__ Wk, const float* __restrict__ bk,
                 const float* __restrict__ Wv, const float* __restrict__ bv,
                 const float* __restrict__ Ws, const float* __restrict__ bs,
                 float* __restrict__ qo, float* __restrict__ ko,
                 float* __restrict__ vo, float* __restrict__ so, int Npad) {
    __shared__ float sW[4 * WSTRIDE];   // swizzled: [(k>>1)*64 + 2*n + (k&1)]
    __shared__ float sB[4 * DHID];
    const int tid = threadIdx.x;
    // zero-fill padding, then scatter real weights into swizzled layout
    for (int i = tid; i < 4 * WSTRIDE; i += 256) sW[i] = 0.0f;
    __syncthreads();
    for (int i = tid; i < DHID * DHID; i += 256) {
        const int kRow = i / DHID;
        const int n = i - kRow * DHID;
        const int off = (kRow >> 1) * (2 * NPADW) + 2 * n + (kRow & 1);
        sW[0 * WSTRIDE + off] = Wq[i];
        sW[1 * WSTRIDE + off] = Wk[i];
        sW[2 * WSTRIDE + off] = Wv[i];
        sW[3 * WSTRIDE + off] = Ws[i];
    }
    if (tid < DHID) {
        sB[0 * DHID + tid] = bq[tid];
        sB[1 * DHID + tid] = bk[tid];
        sB[2 * DHID + tid] = bv[tid];
        sB[3 * DHID + tid] = bs[tid];
    }
    __syncthreads();

    const int lane = tid & 31;
    const int wavesPerBlock = 256 >> 5;
    const int waveId = blockIdx.x * wavesPerBlock + (tid >> 5);
    const int nWaves = gridDim.x * wavesPerBlock;
    const int nTiles = Npad >> 4;

    const int mrow = lane & 15;          // A-matrix row for this lane
    const int khi  = (lane >> 4) << 1;   // lanes 0-15: K=k,k+1; 16-31: K=k+2,k+3
    const int ncol = lane & 15;          // B/D column within 16-wide n-tile

    for (int tile = waveId; tile < nTiles; tile += nWaves) {
        const int rowBase = tile << 4;
        const float* __restrict__ aRow = h + (size_t)(rowBase + mrow) * HPAD;

        v8f acc[4][2] = {};

        #pragma unroll
        for (int kk = 0; kk < 28; kk += 4) {
            const int k0 = kk + khi;                     // even -> 8B aligned
            const v2f a = *(const v2f*)(aRow + k0);      // unconditional global b64
            const int p = (kk >> 1) + (lane >> 4);       // k-pair index
            #pragma unroll
            for (int m = 0; m < 4; ++m) {
                #pragma unroll
                for (int nt = 0; nt < 2; ++nt) {
                    const int n = nt * 16 + ncol;
                    const v2f b = *(const v2f*)&sW[m * WSTRIDE + p * (2 * NPADW) + 2 * n];
                    acc[m][nt] = __builtin_amdgcn_wmma_f32_16x16x4_f32(
                        false, a, false, b, (short)0, acc[m][nt], false, false);
                }
            }
        }

        // D layout: VGPR i holds row i (lanes 0-15) / row i+8 (lanes 16-31), col = lane&15
        const int mOff = (lane >> 4) << 3;
        #pragma unroll
        for (int nt = 0; nt < 2; ++nt) {
            const int col = nt * 16 + ncol;
            if (col < DHID) {
                #pragma unroll
                for (int i = 0; i < 8; ++i) {
                    const int orow = rowBase + mOff + i;
                    qo[(size_t)orow * DHID + col] = acc[0][nt][i] + sB[0 * DHID + col];
                    ko[(size_t)orow * DHID + col] = acc[1][nt][i] + sB[1 * DHID + col];
                    vo[(size_t)orow * DHID + col] = acc[2][nt][i] + sB[2 * DHID + col];
                    so[(size_t)orow * HPAD + col] = acc[3][nt][i] + sB[3 * DHID + col];
                }
            }
        }
    }
}

// ---------- per-layer init: amax keys = 0 (minimal ordered key), denom = 0 ----------
__global__ void __launch_bounds__(256)
init_nh_kernel(unsigned* __restrict__ amaxKey, float* __restrict__ denom, int NH) {
    for (int t = blockIdx.x * 256 + threadIdx.x; t < NH; t += gridDim.x * 256) {
        amaxKey[t] = 0u;
        denom[t] = 0.0f;
    }
}

// ---------- pass A: alpha + scatter segment-max (one thread per edge-head) ----------
__global__ void __launch_bounds__(256)
edge_alpha_kernel(const int* __restrict__ ei, const float* __restrict__ ea,
                  const float* __restrict__ WeL,
                  const float* __restrict__ q, const float* __restrict__ k,
                  float* __restrict__ alphaBuf, unsigned* __restrict__ amaxKey, int E) {
    __shared__ float sWe[2 * DHID];
    if (threadIdx.x < 2 * DHID) sWe[threadIdx.x] = WeL[threadIdx.x];
    __syncthreads();
    const int total = E * HEADS;
    for (int t = blockIdx.x * 256 + threadIdx.x; t < total; t += gridDim.x * 256) {
        const int e  = t / HEADS;
        const int hh = t - e * HEADS;
        const int src = ei[e];
        const int dst = ei[E + e];
        const float ea0 = ea[2 * e], ea1 = ea[2 * e + 1];
        float acc = 0.0f;
        #pragma unroll
        for (int c = 0; c < CH; ++c) {
            const int j = hh * CH + c;
            const float ev = ea0 * sWe[j] + ea1 * sWe[DHID + j];
            acc += q[(size_t)dst * DHID + j] * (k[(size_t)src * DHID + j] + ev);
        }
        const float alpha = acc * 0.44721359549995793f;  // 1/sqrt(C)
        alphaBuf[t] = alpha;
        atomicMax(&amaxKey[dst * HEADS + hh], fkey(alpha));
    }
}

// ---------- pass B: exp + scatter segment-sum ----------
__global__ void __launch_bounds__(256)
edge_exp_kernel(const int* __restrict__ ei, const unsigned* __restrict__ amaxKey,
                float* __restrict__ alphaBuf, float* __restrict__ denom, int E) {
    const int total = E * HEADS;
    for (int t = blockIdx.x * 256 + threadIdx.x; t < total; t += gridDim.x * 256) {
        const int e  = t / HEADS;
        const int hh = t - e * HEADS;
        const int dst = ei[E + e];
        const float amax = fdecode(amaxKey[dst * HEADS + hh]);
        const float ex = __expf(alphaBuf[t] - amax);
        alphaBuf[t] = ex;
        atomicAdd(&denom[dst * HEADS + hh], ex);
    }
}

// ---------- pass C: weighted message scatter-add into hnew (holds skip, stride HPAD) ----------
__global__ void __launch_bounds__(256)
edge_agg_kernel(const int* __restrict__ ei, const float* __restrict__ ea,
                const float* __restrict__ WeL, const float* __restrict__ v,
                const float* __restrict__ alphaBuf, const float* __restrict__ denom,
                float* __restrict__ hnew, int E) {
    __shared__ float sWe[2 * DHID];
    if (threadIdx.x < 2 * DHID) sWe[threadIdx.x] = WeL[threadIdx.x];
    __syncthreads();
    const int total = E * HEADS;
    for (int t = blockIdx.x * 256 + threadIdx.x; t < total; t += gridDim.x * 256) {
        const int e  = t / HEADS;
        const int hh = t - e * HEADS;
        const int src = ei[e];
        const int dst = ei[E + e];
        const float w = alphaBuf[t] / (denom[dst * HEADS + hh] + 1e-16f);
        const float ea0 = ea[2 * e], ea1 = ea[2 * e + 1];
        #pragma unroll
        for (int c = 0; c < CH; ++c) {
            const int j = hh * CH + c;
            const float ev = ea0 * sWe[j] + ea1 * sWe[DHID + j];
            atomicAdd(&hnew[(size_t)dst * HPAD + j], (v[(size_t)src * DHID + j] + ev) * w);
        }
    }
}

// ---------- output head: sigmoid(h @ Wo + bo) ----------
__global__ void __launch_bounds__(256)
out_head_kernel(const float* __restrict__ h, const float* __restrict__ Wo,
                const float* __restrict__ bo, float* __restrict__ out, int N) {
    __shared__ float sWo[DHID];
    __shared__ float sbo;
    if (threadIdx.x < DHID) sWo[threadIdx.x] = Wo[threadIdx.x];
    if (threadIdx.x == 0) sbo = bo[0];
    __syncthreads();
    for (int n = blockIdx.x * 256 + threadIdx.x; n < N; n += gridDim.x * 256) {
        float acc = sbo;
        #pragma unroll
        for (int j = 0; j < DHID; ++j) acc += h[(size_t)n * HPAD + j] * sWo[j];
        out[n] = 1.0f / (1.0f + __expf(-acc));
    }
}

extern "C" void kernel_launch(void* const* d_in, const int* in_sizes, int n_in,
                              void* d_out, int out_size, void* d_ws, size_t ws_size,
                              hipStream_t stream) {
    const float* x     = (const float*)d_in[0];
    const int*   ei    = (const int*)d_in[1];
    const float* ea    = (const float*)d_in[2];
    const float* Wi    = (const float*)d_in[3];
    const float* bi    = (const float*)d_in[4];
    const float* Wq    = (const float*)d_in[5];
    const float* bq    = (const float*)d_in[6];
    const float* Wk    = (const float*)d_in[7];
    const float* bk    = (const float*)d_in[8];
    const float* Wv    = (const float*)d_in[9];
    const float* bv    = (const float*)d_in[10];
    const float* We    = (const float*)d_in[11];
    const float* Wskip = (const float*)d_in[12];
    const float* bskip = (const float*)d_in[13];
    const float* Wo    = (const float*)d_in[14];
    const float* bo    = (const float*)d_in[15];

    const int N = in_sizes[0] / 6;       // x is [N, 6]
    const int E = in_sizes[1] / 2;       // edge_index is [2, E]
    const int Npad = (N + 15) & ~15;     // row-padded so GEMM tiles need no bounds checks

    // workspace carve-up (all float-aligned)
    float* p = (float*)d_ws;
    float* hA       = p; p += (size_t)Npad * HPAD;
    float* hB       = p; p += (size_t)Npad * HPAD;
    float* qb       = p; p += (size_t)Npad * DHID;
    float* kb       = p; p += (size_t)Npad * DHID;
    float* vb       = p; p += (size_t)Npad * DHID;
    float* alphaBuf = p; p += (size_t)E * HEADS;
    unsigned* amaxKey = (unsigned*)p; p += (size_t)N * HEADS;
    float* denom    = p; p += (size_t)N * HEADS;

    // zero h buffers once per call: guarantees K-padding columns and tail rows are 0
    // (async memset is graph-capture safe)
    hipMemsetAsync(hA, 0, (size_t)Npad * HPAD * 2 * sizeof(float), stream);

    const int nodeBlocks = (N + 255) / 256;
    const int nhBlocks   = (N * HEADS + 255) / 256;
    const int edgeBlocks = (int)(((long long)E * HEADS + 255) / 256);
    const int gemmBlocks = ((Npad >> 4) + 7) / 8;   // 8 waves/block, 1 tile/wave

    input_proj_kernel<<<nodeBlocks, 256, 0, stream>>>(x, Wi, bi, hA, N);

    float* hCur = hA;
    float* hNext = hB;
    for (int l = 0; l < 3; ++l) {
        qkvs_wmma_kernel<<<gemmBlocks, 256, 0, stream>>>(
            hCur,
            Wq + l * 625, bq + l * DHID,
            Wk + l * 625, bk + l * DHID,
            Wv + l * 625, bv + l * DHID,
            Wskip + l * 625, bskip + l * DHID,
            qb, kb, vb, hNext, Npad);
        init_nh_kernel<<<nhBlocks, 256, 0, stream>>>(amaxKey, denom, N * HEADS);
        edge_alpha_kernel<<<edgeBlocks, 256, 0, stream>>>(
            ei, ea, We + l * 2 * DHID, qb, kb, alphaBuf, amaxKey, E);
        edge_exp_kernel<<<edgeBlocks, 256, 0, stream>>>(ei, amaxKey, alphaBuf, denom, E);
        edge_agg_kernel<<<edgeBlocks, 256, 0, stream>>>(
            ei, ea, We + l * 2 * DHID, vb, alphaBuf, denom, hNext, E);
        float* tmp = hCur; hCur = hNext; hNext = tmp;
    }

    out_head_kernel<<<nodeBlocks, 256, 0, stream>>>(hCur, Wo, bo, (float*)d_out, N);
}